// BPTTNueron_51659866636869
// MI455X (gfx1250) — compile-verified
//
#include <hip/hip_runtime.h>
#include <cstdint>

// ---------------------------------------------------------------------------
// LIF neuron forward scan, x:[B,N,T] f32 -> spikes:[B,N,T] f32, T contiguous.
// HBM-bound (256 MiB traffic, ~11.5 us floor at 23.3 TB/s on MI455X).
// CDNA5 strategy:
//   * TENSOR_LOAD_TO_LDS (TDM) stages each 128-row x 64-step tile into LDS
//     with ONE scalar-issued DMA per chunk: tile_dim0=64, tile_dim1=128,
//     tensor_dim0_stride=256, and hardware row padding (4 dwords per 64)
//     reproducing the 68-dword conflict-limited LDS layout.
//   * Ping-pong buffers pipelined with s_wait_tensorcnt thresholds
//     (TDM ops complete in order per wave): chunk c+1 streams during scan c.
//   * Tail rows handled by TDM OOB semantics (reads beyond tensor_dim1 -> 0).
//   * Spikes staged in padded LDS, written back with coalesced non-temporal
//     b128 stores (pure streaming, no reuse).
// ---------------------------------------------------------------------------

typedef float        v4f  __attribute__((ext_vector_type(4)));
typedef unsigned int u32x4 __attribute__((ext_vector_type(4)));
typedef int          i32x8 __attribute__((ext_vector_type(8)));
typedef int          i32x4 __attribute__((ext_vector_type(4)));

#define T_LEN   256   // timesteps per row (contiguous axis)
#define TC      64    // timesteps per LDS chunk
#define NCHUNK  (T_LEN / TC)
#define ROWS    128   // rows per workgroup (== blockDim.x), 4 wave32
#define STRIDE  68    // dwords per padded LDS row (64 data + 4 pad = 272B)
#define F4S     16    // float4 store slots per thread per tile

// One TDM descriptor: copy a [rows x 64 float] tile (row stride 256 floats)
// from global to LDS, padding LDS by 4 dwords after every 64 (row end).
__device__ __forceinline__ void tdm_load_tile(uint32_t lds_addr,
                                              const float* gptr,
                                              uint32_t valid_rows) {
    const uint64_t ga = (uint64_t)(uintptr_t)gptr;

    u32x4 g0;
    g0.x = 1u;                                   // count=1 user descriptor
    g0.y = lds_addr;                             // LDS byte address
    g0.z = (uint32_t)ga;                         // global_addr[31:0]
    g0.w = (uint32_t)(ga >> 32) | (2u << 30);    // global_addr[56:32], type=2

    i32x8 g1;
    g1[0] = (int)((2u << 16)      // data_size = 4 bytes
                | (1u << 20)      // pad_enable
                | (5u << 22)      // pad_interval code 5 = every 64 dwords
                | (3u << 25));    // pad_amount  code 3 = 4 dwords
    g1[1] = (int)(64u << 16);                    // tensor_dim0 = 64 (bits 79:48)
    g1[2] = (int)((valid_rows & 0xFFFFu) << 16); // tensor_dim1 (bits 111:80)
    g1[3] = (int)(64u << 16);                    // tile_dim0 = 64 (bits 127:112)
    g1[4] = (int)ROWS;                           // tile_dim1 = 128 (bits 143:128)
    g1[5] = (int)T_LEN;                          // tensor_dim0_stride = 256
    g1[6] = 0;                                   // tensor_dim1_stride = 0 (2D)
    g1[7] = 0;

    const i32x4 gz4 = {0, 0, 0, 0};              // groups 2/3 unused (2D)
    const i32x8 gz8 = {0, 0, 0, 0, 0, 0, 0, 0};  // extra group (6-arg form)
    __builtin_amdgcn_tensor_load_to_lds(g0, g1, gz4, gz4, gz8, 0);
}

template <bool FULL>
__device__ __forceinline__ void run_block(const float* __restrict__ x,
                                          float* __restrict__ out,
                                          long long rowBase, int nrows,
                                          float (&in_tile)[2][ROWS * STRIDE],
                                          float (&out_tile)[ROWS * STRIDE]) {
    const int tid = threadIdx.x;
    const int r0  = tid >> 4;   // first local row this thread stores (0..7)
    const int i4  = tid & 15;   // float4 index within a row-chunk (0..15)

    const uint32_t lds_base[2] = {
        (uint32_t)(uintptr_t)(&in_tile[0][0]),   // low 32 bits == LDS address
        (uint32_t)(uintptr_t)(&in_tile[1][0])
    };
    const float* xbase = x + rowBase * T_LEN;    // block-uniform
    const uint32_t vrows = FULL ? (uint32_t)ROWS
                                : (uint32_t)(nrows - rowBase);  // OOB rows -> 0
    const bool issuer = (tid < 32);              // exactly one wave drives TDM

    float u = 0.0f;  // membrane potential, persists across chunks (per row)

    if (issuer) tdm_load_tile(lds_base[0], xbase, vrows);   // prologue DMA

    #pragma unroll
    for (int c = 0; c < NCHUNK; ++c) {
        const int buf = c & 1;

        if (issuer) {
            // Prefetch next chunk into the other buffer (last read in
            // compute(c-1); ordered by the end-of-iteration barrier), then
            // wait for chunk c only (TDM completes in order per wave).
            if (c + 1 < NCHUNK) {
                tdm_load_tile(lds_base[buf ^ 1], xbase + (c + 1) * TC, vrows);
                __builtin_amdgcn_s_wait_tensorcnt(1);
            } else {
                __builtin_amdgcn_s_wait_tensorcnt(0);
            }
        }
        __syncthreads();   // publish chunk c's tile to all waves

        // ---------------- sequential LIF scan out of LDS ------------------
        {
            const int rbase = tid * STRIDE;
            #pragma unroll
            for (int j = 0; j < TC / 4; ++j) {
                const v4f xv =
                    *reinterpret_cast<const v4f*>(&in_tile[buf][rbase + j * 4]);
                v4f sv;

                u = __builtin_fmaf(u, 0.125f, xv.x);
                { const bool f = (u >= 1.0f); sv.x = f ? 1.0f : 0.0f; u = f ? 0.0f : u; }
                u = __builtin_fmaf(u, 0.125f, xv.y);
                { const bool f = (u >= 1.0f); sv.y = f ? 1.0f : 0.0f; u = f ? 0.0f : u; }
                u = __builtin_fmaf(u, 0.125f, xv.z);
                { const bool f = (u >= 1.0f); sv.z = f ? 1.0f : 0.0f; u = f ? 0.0f : u; }
                u = __builtin_fmaf(u, 0.125f, xv.w);
                { const bool f = (u >= 1.0f); sv.w = f ? 1.0f : 0.0f; u = f ? 0.0f : u; }

                *reinterpret_cast<v4f*>(&out_tile[rbase + j * 4]) = sv;
            }
        }
        __syncthreads();   // spike tile complete

        // ---------------- coalesced non-temporal spike stores -------------
        {
            float* obase = out + (rowBase + r0) * T_LEN + c * TC + i4 * 4;
            const int lbase = r0 * STRIDE + i4 * 4;
            #pragma unroll
            for (int it = 0; it < F4S; ++it) {
                if (FULL || rowBase + (r0 + it * 8) < nrows) {
                    const v4f v = *reinterpret_cast<const v4f*>(
                        &out_tile[lbase + it * 8 * STRIDE]);
                    __builtin_nontemporal_store(
                        v, reinterpret_cast<v4f*>(obase + it * 8 * T_LEN));
                }
            }
        }
        __syncthreads();   // out_tile + the prefetch target safe to reuse
    }
}

__global__ __launch_bounds__(ROWS)
void lif_scan_kernel(const float* __restrict__ x,
                     float* __restrict__ out,
                     int nrows) {
    __shared__ __align__(16) float in_tile[2][ROWS * STRIDE];
    __shared__ __align__(16) float out_tile[ROWS * STRIDE];

    const long long rowBase = (long long)blockIdx.x * ROWS;
    if (rowBase + ROWS <= (long long)nrows) {
        run_block<true >(x, out, rowBase, nrows, in_tile, out_tile);  // hot
    } else {
        run_block<false>(x, out, rowBase, nrows, in_tile, out_tile);  // tail
    }
}

extern "C" void kernel_launch(void* const* d_in, const int* in_sizes, int n_in,
                              void* d_out, int out_size, void* d_ws, size_t ws_size,
                              hipStream_t stream) {
    (void)n_in; (void)d_ws; (void)ws_size; (void)out_size;
    const float* x = (const float*)d_in[0];
    float* out = (float*)d_out;

    const int total = in_sizes[0];          // B*N*T = 33,554,432
    const int nrows = total / T_LEN;        // 131,072
    const int nblocks = (nrows + ROWS - 1) / ROWS;   // 1024

    lif_scan_kernel<<<nblocks, ROWS, 0, stream>>>(x, out, nrows);
}